// SubqueryFeatureSampler_77051713290515
// MI455X (gfx1250) — compile-verified
//
#include <hip/hip_runtime.h>
#include <hip/hip_bf16.h>

// ---------------- problem constants ----------------
constexpr int B_ = 2, N_ = 100, C_ = 256, G_ = 4, P_ = 32, POOL_ = 9;
constexpr int Cg_ = C_ / G_;          // 64
constexpr int S_ = P_ * POOL_;        // 288
constexpr float TAU_ = 2.0f;
constexpr int BN_ = B_ * N_;          // 200
constexpr int NC1_ = 3 * G_ * P_;     // 384
constexpr int NC2_ = 3 * POOL_;       // 27

typedef __attribute__((ext_vector_type(2))) float v2f;
typedef __attribute__((ext_vector_type(8))) float v8f;

// ---------------------------------------------------------------------------
// GEMM: out[M,Ncols] = X[M,K] @ Wm[K,Ncols] + bias  using V_WMMA_F32_16X16X4_F32
// One wave (32 threads) per 16x16 output tile, K stepped by 4.
// A-frag: lane holds row M=lane&15, k-pair (lane>>4)*2  (2 VGPRs)
// B-frag: lane holds col N=lane&15, k-pair (lane>>4)*2  (2 VGPRs)
// C/D:    VGPR r <-> M = r + 8*(lane>>4), N = lane&15   (8 VGPRs)
// ---------------------------------------------------------------------------
__global__ void gemm_wmma_f32_kernel(const float* __restrict__ X,
                                     const float* __restrict__ Wm,
                                     const float* __restrict__ bias,
                                     float* __restrict__ out,
                                     int M, int K, int Ncols, int tilesN)
{
    const int wid  = blockIdx.x;
    const int tm   = wid / tilesN;
    const int tn   = wid % tilesN;
    const int lane = threadIdx.x;          // 0..31 (wave32)
    const int l15  = lane & 15;
    const int khalf = (lane >> 4) << 1;    // 0 or 2

    const int mrow = tm * 16 + l15;
    const int mc   = min(mrow, M - 1);     // clamp; masked at store
    const int col  = tn * 16 + l15;
    const int cc   = min(col, Ncols - 1);  // clamp; masked at store

    const float* __restrict__ xrow = X + (size_t)mc * K;

    v8f acc = {0.f, 0.f, 0.f, 0.f, 0.f, 0.f, 0.f, 0.f};
    for (int k = 0; k < K; k += 4) {
        const int ka = k + khalf;
        v2f a, b;
        a.x = xrow[ka];
        a.y = xrow[ka + 1];
        const float* wp = Wm + (size_t)ka * Ncols + cc;
        b.x = wp[0];
        b.y = wp[Ncols];
        acc = __builtin_amdgcn_wmma_f32_16x16x4_f32(
            /*neg_a=*/false, a, /*neg_b=*/false, b,
            /*c_mod=*/(short)0, acc, /*reuse_a=*/false, /*reuse_b=*/false);
    }

    const float bv    = bias[cc];
    const int   rbase = tm * 16 + ((lane >> 4) << 3);
#pragma unroll
    for (int r = 0; r < 8; ++r) {
        const int mr = rbase + r;
        if (mr < M && col < Ncols)
            out[(size_t)mr * Ncols + col] = acc[r] + bv;
    }
}

// ---------------------------------------------------------------------------
// Sampler: one thread per (b,n,g,p,q) point. Computes the sampling position,
// the 4-level softmax z-weights, folds level weight * bilinear corner weight *
// zero-padding validity into 16 (level,corner) coefficients, then gathers
// 64 channels (L2-resident pyramid) and writes 256 contiguous bytes.
// ---------------------------------------------------------------------------
__global__ void __launch_bounds__(256)
sampler_kernel(const float* __restrict__ f0, const float* __restrict__ f1,
               const float* __restrict__ f2, const float* __restrict__ f3,
               const float* __restrict__ qbox,
               const float* __restrict__ d00,   // [BN, 384]
               const float* __restrict__ d01,   // [BN, 27]
               float* __restrict__ out)         // [B,N,G,S,Cg]
{
    const int t = blockIdx.x * blockDim.x + threadIdx.x;
    constexpr int TOT = B_ * N_ * G_ * S_;     // 230400
    if (t >= TOT) return;

    // t == (((b*N+n)*G+g)*S + s) with s = p*POOL + q  (q fastest)
    int tmp = t;
    const int q = tmp % POOL_;  tmp /= POOL_;
    const int p = tmp % P_;     tmp /= P_;
    const int g = tmp % G_;     tmp /= G_;
    const int n = tmp % N_;
    const int b = tmp / N_;
    const int bn = b * N_ + n;

    const float* d0 = d00 + ((size_t)bn * (G_ * P_) + g * P_ + p) * 3;
    const float dx0 = d0[0], dy0 = d0[1], dz0 = d0[2];
    const float* d1 = d01 + ((size_t)bn * POOL_ + q) * 3;
    const float dx1 = d1[0], dy1 = d1[1], dz1 = d1[2];
    const float* qb = qbox + (size_t)bn * 4;
    const float ox = qb[0], oy = qb[1], oz = qb[2], orr = qb[3];

    const float sz0  = exp2f(dz0);
    const float newx = ox + exp2f(oz - 0.5f * orr) * (dx0 + sz0 * dx1);
    const float newy = oy + exp2f(oz + 0.5f * orr) * (dy0 + sz0 * dy1);
    const float newz = oz + dz0 + dz1;

    // softmax over 4 levels of -(z-l)^2 / TAU
    float wz[4];
    float mx = -1e30f;
#pragma unroll
    for (int l = 0; l < 4; ++l) {
        const float d = newz - (float)l;
        wz[l] = -d * d * (1.0f / TAU_);
        mx = fmaxf(mx, wz[l]);
    }
    float sum = 0.f;
#pragma unroll
    for (int l = 0; l < 4; ++l) { wz[l] = __expf(wz[l] - mx); sum += wz[l]; }
    const float rsum = 1.0f / sum;

    const float* fms[4] = {f0, f1, f2, f3};
    int   idx[4][4];
    float wgt[4][4];
    int   hw[4];
    const float* basep[4];

#pragma unroll
    for (int l = 0; l < 4; ++l) {
        const int   Hl   = 128 >> l;          // 128,64,32,16
        const int   Wl   = Hl;
        const float strd = (float)(8 << l);   // 8,16,32,64
        // W*stride == 1024 for all levels, so grid_sample reduces to:
        const float xs = newx / strd - 0.5f;
        const float ys = newy / strd - 0.5f;
        const float xf = floorf(xs), yf = floorf(ys);
        const int   x0 = (int)xf,   y0 = (int)yf;
        const float wx1 = xs - xf,  wy1 = ys - yf;
        const float wx0 = 1.f - wx1, wy0 = 1.f - wy1;
        const int   x1 = x0 + 1,    y1 = y0 + 1;
        const bool vx0 = (x0 >= 0) & (x0 < Wl);
        const bool vx1 = (x1 >= 0) & (x1 < Wl);
        const bool vy0 = (y0 >= 0) & (y0 < Hl);
        const bool vy1 = (y1 >= 0) & (y1 < Hl);
        const int x0c = min(max(x0, 0), Wl - 1), x1c = min(max(x1, 0), Wl - 1);
        const int y0c = min(max(y0, 0), Hl - 1), y1c = min(max(y1, 0), Hl - 1);
        const float lw = wz[l] * rsum;
        wgt[l][0] = lw * wx0 * wy0 * (float)(vx0 && vy0);
        wgt[l][1] = lw * wx1 * wy0 * (float)(vx1 && vy0);
        wgt[l][2] = lw * wx0 * wy1 * (float)(vx0 && vy1);
        wgt[l][3] = lw * wx1 * wy1 * (float)(vx1 && vy1);
        idx[l][0] = y0c * Wl + x0c;
        idx[l][1] = y0c * Wl + x1c;
        idx[l][2] = y1c * Wl + x0c;
        idx[l][3] = y1c * Wl + x1c;
        hw[l]     = Hl * Wl;
        basep[l]  = fms[l] + (size_t)(b * C_ + g * Cg_) * hw[l];
    }

    float* __restrict__ op = out + (size_t)t * Cg_;

#pragma unroll 4
    for (int c = 0; c < Cg_; c += 4) {
        float a0 = 0.f, a1 = 0.f, a2 = 0.f, a3 = 0.f;
#pragma unroll
        for (int l = 0; l < 4; ++l) {
            const int h = hw[l];
            const float* __restrict__ fp = basep[l] + (size_t)c * h;
#pragma unroll
            for (int k = 0; k < 4; ++k) {
                const float w_ = wgt[l][k];
                const int   id = idx[l][k];
                a0 += w_ * fp[id];
                a1 += w_ * fp[id + h];
                a2 += w_ * fp[id + 2 * h];
                a3 += w_ * fp[id + 3 * h];
            }
        }
        float4 v = make_float4(a0, a1, a2, a3);
        *(float4*)(op + c) = v;
    }
}

extern "C" void kernel_launch(void* const* d_in, const int* in_sizes, int n_in,
                              void* d_out, int out_size, void* d_ws, size_t ws_size,
                              hipStream_t stream) {
    const float* f0       = (const float*)d_in[0];
    const float* f1       = (const float*)d_in[1];
    const float* f2       = (const float*)d_in[2];
    const float* f3       = (const float*)d_in[3];
    const float* qc       = (const float*)d_in[4];
    const float* qbox     = (const float*)d_in[5];
    const float* W_one    = (const float*)d_in[6];
    const float* b_one    = (const float*)d_in[7];
    const float* W_double = (const float*)d_in[8];
    const float* b_double = (const float*)d_in[9];
    float* out = (float*)d_out;

    float* d00 = (float*)d_ws;                    // BN*384 floats
    float* d01 = d00 + (size_t)BN_ * NC1_;        // BN*27 floats

    // GEMM 1: [200,256]x[256,384] -> 13x24 tiles of 16x16, one wave each
    {
        const int tilesM = (BN_ + 15) / 16;       // 13
        const int tilesN = NC1_ / 16;             // 24
        gemm_wmma_f32_kernel<<<tilesM * tilesN, 32, 0, stream>>>(
            qc, W_one, b_one, d00, BN_, C_, NC1_, tilesN);
    }
    // GEMM 2: [200,256]x[256,27] -> 13x2 tiles
    {
        const int tilesM = (BN_ + 15) / 16;       // 13
        const int tilesN = (NC2_ + 15) / 16;      // 2
        gemm_wmma_f32_kernel<<<tilesM * tilesN, 32, 0, stream>>>(
            qc, W_double, b_double, d01, BN_, C_, NC2_, tilesN);
    }
    // Sampler: one thread per (b,n,g,p,q) point
    {
        constexpr int TOT = B_ * N_ * G_ * S_;    // 230400
        const int blk = 256;
        const int grd = (TOT + blk - 1) / blk;    // 900
        sampler_kernel<<<grd, blk, 0, stream>>>(f0, f1, f2, f3, qbox, d00, d01, out);
    }
}